// SpikingEmbedding_84155589198552
// MI455X (gfx1250) — compile-verified
//
#include <hip/hip_runtime.h>
#include <cstdint>

// Problem shape (from reference): ids [8,512] i32, table [50257,512] f32,
// out [8,16,512,512] f32 spikes.
#define BATCH 8
#define SEQ   512
#define DIM   512
#define TSTEPS 16
#define DECAY 0.9f
#define THRESH 1.0f

typedef uint32_t u32;
typedef __attribute__((ext_vector_type(4))) unsigned int v4u32;
typedef __attribute__((ext_vector_type(8))) int v8i32;
typedef __attribute__((ext_vector_type(4))) int v4i32;
typedef __attribute__((ext_vector_type(4))) float v4f;   // native clang vector (NT-store OK)

__global__ __launch_bounds__(128) void SpikingEmbedding_kernel(
    const int* __restrict__ ids,
    const float* __restrict__ table,
    float* __restrict__ out) {
  // One block per (b, s) row.
  __shared__ __align__(16) float row[DIM];

  const u32 r = blockIdx.x;            // 0 .. BATCH*SEQ-1
  const int id = ids[r];               // uniform -> scalar load

  // ---- TDM: DMA one 2KB embedding row (global -> LDS), wave 0 only ----
  if (threadIdx.x < 32) {
    const uint64_t gaddr =
        (uint64_t)(uintptr_t)table + (uint64_t)(u32)id * (DIM * 4ull);
    const u32 lds_base = (u32)(uintptr_t)(void*)&row[0];  // low 32 bits = LDS offset

    // D# group 0 (128b): count=1, lds_addr, global_addr[56:0], type=2
    v4u32 g0;
    g0.x = 1u;                                        // count=1, user desc
    g0.y = lds_base;                                  // lds_addr (bytes)
    g0.z = (u32)(gaddr & 0xFFFFFFFFull);              // global_addr[31:0]
    g0.w = (u32)((gaddr >> 32) & 0x01FFFFFFull)       // global_addr[56:32]
         | (2u << 30);                                // type = 2 ("image")

    // D# group 1 (256b): data_size=4B, tensor_dim0=512, tensor_dim1=1,
    // tile = 512 x 1, tensor_dim0_stride = 512 elements. No pad/iterate/mask.
    v8i32 g1;
    g1[0] = (int)(2u << 16);                          // workgroup_mask=0, data_size=2 (4B)
    g1[1] = (int)((u32)(DIM & 0xFFFF) << 16);         // tensor_dim0[15:0] in [31:16]
    g1[2] = (int)(1u << 16);                          // tensor_dim0[31:16]=0 | tensor_dim1=1
    g1[3] = (int)((u32)(DIM & 0xFFFF) << 16);         // tensor_dim1 hi=0 | tile_dim0=512
    g1[4] = 1;                                        // tile_dim1=1, tile_dim2=0
    g1[5] = DIM;                                      // tensor_dim0_stride[31:0]
    g1[6] = 0;                                        // stride hi / dim1_stride lo
    g1[7] = 0;                                        // dim1_stride hi

    v4i32 z4 = {0, 0, 0, 0};                          // groups 2/3: dims unused
#if defined(__clang_major__) && __clang_major__ >= 23
    v8i32 z8 = {0, 0, 0, 0, 0, 0, 0, 0};
    __builtin_amdgcn_tensor_load_to_lds(g0, g1, z4, z4, z8, 0);
#else
    __builtin_amdgcn_tensor_load_to_lds(g0, g1, z4, z4, 0);
#endif
    __builtin_amdgcn_s_wait_tensorcnt(0);
  }
  __syncthreads();

  // ---- LIF scan: each thread owns 4 consecutive d values ----
  const u32 d0 = threadIdx.x * 4u;
  const v4f e = *reinterpret_cast<const v4f*>(&row[d0]);

  // rate = sigmoid(e)
  const float r0 = 1.0f / (1.0f + __expf(-e.x));
  const float r1 = 1.0f / (1.0f + __expf(-e.y));
  const float r2 = 1.0f / (1.0f + __expf(-e.z));
  const float r3 = 1.0f / (1.0f + __expf(-e.w));

  const u32 b = r >> 9;                // r / SEQ
  const u32 s = r & (SEQ - 1);         // r % SEQ
  // out[((b*T + t)*S + s)*D + d]
  float* outp = out + (size_t)b * (TSTEPS * SEQ * DIM)
                    + (size_t)s * DIM + d0;

  float v0 = 0.f, v1 = 0.f, v2 = 0.f, v3 = 0.f;
#pragma unroll
  for (int t = 0; t < TSTEPS; ++t) {
    v0 = fmaf(DECAY, v0, r0);
    v1 = fmaf(DECAY, v1, r1);
    v2 = fmaf(DECAY, v2, r2);
    v3 = fmaf(DECAY, v3, r3);
    const float s0 = (v0 >= THRESH) ? 1.0f : 0.0f;
    const float s1 = (v1 >= THRESH) ? 1.0f : 0.0f;
    const float s2 = (v2 >= THRESH) ? 1.0f : 0.0f;
    const float s3 = (v3 >= THRESH) ? 1.0f : 0.0f;
    v0 -= s0 * THRESH;
    v1 -= s1 * THRESH;
    v2 -= s2 * THRESH;
    v3 -= s3 * THRESH;
    v4f o;
    o.x = s0; o.y = s1; o.z = s2; o.w = s3;
    // 128 MB streaming output: bypass near caches (NT store hint)
    __builtin_nontemporal_store(o, reinterpret_cast<v4f*>(outp));
    outp += SEQ * DIM;                 // advance t
  }
}

extern "C" void kernel_launch(void* const* d_in, const int* in_sizes, int n_in,
                              void* d_out, int out_size, void* d_ws, size_t ws_size,
                              hipStream_t stream) {
  const int* ids = (const int*)d_in[0];       // [8, 512] int32
  const float* table = (const float*)d_in[1]; // [50257, 512] f32
  float* out = (float*)d_out;                 // [8, 16, 512, 512] f32

  dim3 grid(BATCH * SEQ);   // 4096 rows, one block each
  dim3 block(128);          // 4 wave32s; 4 d-values per thread
  hipLaunchKernelGGL(SpikingEmbedding_kernel, grid, block, 0, stream,
                     ids, table, out);
}